// TaskAlignedAssigner_17162689314989
// MI455X (gfx1250) — compile-verified
//
#include <hip/hip_runtime.h>
#include <stdint.h>

// Problem constants (match reference setup_inputs)
#define B_       16
#define N_       8400
#define M_       128
#define TOPK_    10
#define EPS_     1e-9f
#define IOU_EPS_ 1e-7f

#define COLS_PER_BLOCK_ 8      // 8 waves per block, one GT column each
#define TILE_           512    // pred boxes staged per LDS tile
#define NT_             ((N_ + TILE_ - 1) / TILE_)   // 17 tiles

// ---------------------------------------------------------------------------
// wave32 max-reduction of a 64-bit key (two 32-bit shuffles per step)
// ---------------------------------------------------------------------------
static __device__ __forceinline__ unsigned long long wave_max_u64(unsigned long long v) {
#pragma unroll
  for (int m = 16; m > 0; m >>= 1) {
    unsigned lo  = (unsigned)v;
    unsigned hi  = (unsigned)(v >> 32);
    unsigned olo = (unsigned)__shfl_xor((int)lo, m, 32);
    unsigned ohi = (unsigned)__shfl_xor((int)hi, m, 32);
    unsigned long long o = ((unsigned long long)ohi << 32) | (unsigned long long)olo;
    if (o > v) v = o;
  }
  return v;
}

// ---------------------------------------------------------------------------
// Kernel 0: zero the per-(b,n) packed-key workspace
// ---------------------------------------------------------------------------
__global__ void taa_init_ws(unsigned long long* __restrict__ keys) {
  int i = blockIdx.x * blockDim.x + threadIdx.x;
  if (i < B_ * N_) keys[i] = 0ull;
}

// ---------------------------------------------------------------------------
// Async-stage one pred tile (TILE_ boxes + scores) into LDS.
// All 256 threads participate; indices are clamped inside the batch so every
// wave issues exactly 3 async instructions -> s_wait_asynccnt 3 is precise.
// ---------------------------------------------------------------------------
static __device__ __forceinline__ void stage_tile(
    const void* box_src,   // batch base of pred_boxes (float4*)
    const void* sc_src,    // batch base of pred_scores (float*)
    int tile_base, int tid, float4* sb, float* ss) {
  int i0 = tile_base + tid;
  int i1 = tile_base + 256 + tid;
  i0 = (i0 < N_) ? i0 : (N_ - 1);
  i1 = (i1 < N_) ? i1 : (N_ - 1);
  unsigned lds0 = (unsigned)(uintptr_t)(void*)&sb[tid];
  unsigned lds1 = (unsigned)(uintptr_t)(void*)&sb[256 + tid];
  unsigned o0   = (unsigned)i0 * 16u;
  unsigned o1   = (unsigned)i1 * 16u;
  asm volatile("global_load_async_to_lds_b128 %0, %1, %2"
               :: "v"(lds0), "v"(o0), "s"(box_src) : "memory");
  asm volatile("global_load_async_to_lds_b128 %0, %1, %2"
               :: "v"(lds1), "v"(o1), "s"(box_src) : "memory");
  int s0 = tile_base + tid * 2;
  s0 = (s0 <= N_ - 2) ? s0 : (N_ - 2);
  unsigned ldss = (unsigned)(uintptr_t)(void*)&ss[tid * 2];
  unsigned os   = (unsigned)s0 * 4u;
  asm volatile("global_load_async_to_lds_b64 %0, %1, %2"
               :: "v"(ldss), "v"(os), "s"(sc_src) : "memory");
}

// ---------------------------------------------------------------------------
// Kernel 1: 256 threads = 8 waves per block; wave w owns GT column (base+w).
//   Pred boxes/scores are staged tile-by-tile into double-buffered LDS with
//   the gfx1250 async global->LDS path (ASYNCcnt), giving 8-way reuse per
//   load. Each wave keeps a per-lane sorted top-10 of packed keys
//   (metric_bits<<32 | (0xFFFFFFFF-n)), merges lanes with wave32 ballot +
//   shuffle, and scatters winners with a u64 atomic-max whose key
//   (metric_bits<<32 | (M-1-m)) reproduces the reference argmax tie-break.
// ---------------------------------------------------------------------------
__global__ __launch_bounds__(256) void taa_topk_scatter(
    const float*  __restrict__ scores,   // (B,N,1)
    const float4* __restrict__ boxes,    // (B,N,4)
    const float4* __restrict__ gboxes,   // (B,M,4)
    const int*    __restrict__ gmask,    // (B,M)
    unsigned long long* __restrict__ keys) {
  __shared__ float4 sbox[2][TILE_];      // 16 KB
  __shared__ float  ssc [2][TILE_];      //  4 KB

  const int tid  = threadIdx.x;
  const int lane = tid & 31;
  const int w    = tid >> 5;
  const int col  = blockIdx.x * COLS_PER_BLOCK_ + w;   // 8 cols, same batch
  const int b    = col / M_;
  const int m    = col % M_;

  const bool active = (gmask[b * M_ + m] > 0);

  float4 g = make_float4(0.f, 0.f, 0.f, 0.f);
  float  garea = IOU_EPS_;
  if (active) {
    g = gboxes[b * M_ + m];
    garea = fmaxf((g.z - g.x) * (g.w - g.y), IOU_EPS_);
  }

  const void* box_src = (const void*)(boxes  + (size_t)b * N_);
  const void* sc_src  = (const void*)(scores + (size_t)b * N_);

  unsigned long long t[TOPK_];           // ascending sorted top-10 (registers)
#pragma unroll
  for (int j = 0; j < TOPK_; ++j) t[j] = 0ull;

  // -------- software-pipelined tile loop (double-buffered async DMA) -------
  stage_tile(box_src, sc_src, 0, tid, sbox[0], ssc[0]);
  for (int k = 0; k < NT_; ++k) {
    const int cur = k & 1;
    if (k + 1 < NT_) {
      stage_tile(box_src, sc_src, (k + 1) * TILE_, tid, sbox[cur ^ 1], ssc[cur ^ 1]);
      asm volatile("s_wait_asynccnt 3" ::: "memory");   // tile k complete
    } else {
      asm volatile("s_wait_asynccnt 0" ::: "memory");
    }
    __syncthreads();                                    // tile k visible to all

    if (active) {
      const int base = k * TILE_;
      const int cnt  = (N_ - base < TILE_) ? (N_ - base) : TILE_;
      for (int i = lane; i < cnt; i += 32) {
        const float4 p  = sbox[cur][i];
        const float  sc = ssc[cur][i];

        const float parea = fmaxf((p.z - p.x) * (p.w - p.y), IOU_EPS_);
        const float iw    = fminf(p.z, g.z) - fmaxf(p.x, g.x);
        const float ih    = fminf(p.w, g.w) - fmaxf(p.y, g.y);
        const float inter = fmaxf(iw, 0.0f) * fmaxf(ih, 0.0f);
        const float iou   = inter / (parea + garea - inter + IOU_EPS_);
        const float ic    = fminf(fmaxf(iou, 0.0f), 1.0f);
        const float s     = fminf(fmaxf(sc, EPS_), 1.0f);
        const float i2    = ic * ic;
        const float met   = s * (i2 * i2 * i2);   // s^1 * iou^6

        // metric <= EPS (or NaN) can never become foreground: skip (exact,
        // since met > 1e-9 && s <= 1  =>  iou > 10^-1.5 >> EPS).
        if (!(met > EPS_)) continue;

        const int n = base + i;
        const unsigned long long key =
            ((unsigned long long)__float_as_uint(met) << 32) |
            (unsigned long long)(0xFFFFFFFFu - (unsigned)n);  // ties -> smaller n

        if (key > t[0]) {              // branch-free sorted insert
#pragma unroll
          for (int j = 0; j < TOPK_ - 1; ++j)
            t[j] = (key > t[j + 1]) ? t[j + 1] : ((key > t[j]) ? key : t[j]);
          t[TOPK_ - 1] = (key > t[TOPK_ - 1]) ? key : t[TOPK_ - 1];
        }
      }
    }
    __syncthreads();                   // all waves done reading buf[cur]
  }

  // -------- merge 32 per-lane top-10 lists; scatter winners ----------------
  if (!active) return;
  const unsigned long long sec = (unsigned long long)(unsigned)(M_ - 1 - m);
  for (int r = 0; r < TOPK_; ++r) {
    unsigned long long lmax = 0ull;
#pragma unroll
    for (int j = 0; j < TOPK_; ++j) lmax = (t[j] > lmax) ? t[j] : lmax;

    const unsigned long long wmax = wave_max_u64(lmax);
    const float met = __uint_as_float((unsigned)(wmax >> 32));
    if (!(met > EPS_)) break;          // uniform across the wave

    const unsigned long long bal = __ballot(lmax == wmax);   // wave32 low bits
    if (lane == (int)(__ffsll((unsigned long long)bal) - 1)) {
#pragma unroll
      for (int j = 0; j < TOPK_; ++j) if (t[j] == wmax) t[j] = 0ull;  // pop
      const unsigned n = 0xFFFFFFFFu - (unsigned)(wmax & 0xFFFFFFFFull);
      atomicMax(&keys[(size_t)b * N_ + n],
                (wmax & 0xFFFFFFFF00000000ull) | sec);
    }
  }
}

// ---------------------------------------------------------------------------
// Kernel 2: per (b,n) resolve + output write. Stages the batch's GT boxes
// and labels (2.5 KB) into LDS with async global->LDS, then decodes the
// packed key, recomputes the winning IoU and writes all five outputs.
// ---------------------------------------------------------------------------
__global__ __launch_bounds__(256) void taa_resolve(
    const float4* __restrict__ boxes,    // (B,N,4)
    const float4* __restrict__ gboxes,   // (B,M,4)
    const int*    __restrict__ glabels,  // (B,M)
    const unsigned long long* __restrict__ keys,
    int*           __restrict__ o_idx,   // (B,N) int32
    int*           __restrict__ o_lab,   // (B,N) int32
    float4*        __restrict__ o_box,   // (B,N,4) f32
    float*         __restrict__ o_sco,   // (B,N,1) f32
    unsigned char* __restrict__ o_fg) {  // (B,N) bool
  __shared__ float4 sg[M_];
  __shared__ int    sl[M_];

  const int b   = blockIdx.y;
  const int tid = threadIdx.x;

  if (tid < M_) {
    unsigned    lds_b = (unsigned)(uintptr_t)(void*)&sg[tid];
    unsigned    off_b = (unsigned)(tid * 16);
    const void* src_b = (const void*)((const char*)gboxes + (size_t)b * M_ * 16);
    asm volatile("global_load_async_to_lds_b128 %0, %1, %2"
                 :: "v"(lds_b), "v"(off_b), "s"(src_b) : "memory");

    unsigned    lds_l = (unsigned)(uintptr_t)(void*)&sl[tid];
    unsigned    off_l = (unsigned)(tid * 4);
    const void* src_l = (const void*)((const char*)glabels + (size_t)b * M_ * 4);
    asm volatile("global_load_async_to_lds_b32 %0, %1, %2"
                 :: "v"(lds_l), "v"(off_l), "s"(src_l) : "memory");
  }
  asm volatile("s_wait_asynccnt 0" ::: "memory");
  __syncthreads();

  const int n = blockIdx.x * 256 + tid;
  if (n >= N_) return;

  const size_t off = (size_t)b * N_ + n;
  const unsigned long long key = keys[off];
  const float met = __uint_as_float((unsigned)(key >> 32));
  const bool  fg  = met > EPS_;

  int    idx = -1, lab = 0;
  float  iou = 0.0f;
  float4 ob  = make_float4(0.f, 0.f, 0.f, 0.f);

  if (fg) {
    const int m = M_ - 1 - (int)(unsigned)(key & 0xFFFFFFFFull);
    const float4 p = boxes[off];
    const float4 gt = sg[m];
    const float parea = fmaxf((p.z - p.x) * (p.w - p.y), IOU_EPS_);
    const float garea = fmaxf((gt.z - gt.x) * (gt.w - gt.y), IOU_EPS_);
    const float iw    = fminf(p.z, gt.z) - fmaxf(p.x, gt.x);
    const float ih    = fminf(p.w, gt.w) - fmaxf(p.y, gt.y);
    const float inter = fmaxf(iw, 0.0f) * fmaxf(ih, 0.0f);
    iou = inter / (parea + garea - inter + IOU_EPS_);
    idx = m;
    lab = sl[m];
    ob  = gt;
  }

  o_idx[off] = idx;
  o_lab[off] = lab;
  o_box[off] = ob;
  o_sco[off] = iou;
  o_fg [off] = fg ? 1 : 0;
}

// ---------------------------------------------------------------------------
// Host launcher
// ---------------------------------------------------------------------------
extern "C" void kernel_launch(void* const* d_in, const int* in_sizes, int n_in,
                              void* d_out, int out_size, void* d_ws, size_t ws_size,
                              hipStream_t stream) {
  (void)in_sizes; (void)n_in; (void)out_size; (void)ws_size;

  const float* pred_scores = (const float*)d_in[0];   // (B,N,1)
  const float* pred_boxes  = (const float*)d_in[1];   // (B,N,4)
  const int*   gt_labels   = (const int*)  d_in[2];   // (B,M)
  const float* gt_boxes    = (const float*)d_in[3];   // (B,M,4)
  const int*   gt_mask     = (const int*)  d_in[4];   // (B,M)

  unsigned long long* keys = (unsigned long long*)d_ws;   // B*N u64 = 1.05 MB

  // Output tuple, concatenated flat in return order:
  //   idx(B,N,i32) | labels(B,N,i32) | boxes(B,N,4,f32) | scores(B,N,1,f32) | fg(B,N,bool)
  const size_t SZ = (size_t)B_ * N_;
  char* base = (char*)d_out;
  int*           o_idx = (int*)          (base);
  int*           o_lab = (int*)          (base + 4  * SZ);
  float4*        o_box = (float4*)       (base + 8  * SZ);
  float*         o_sco = (float*)        (base + 24 * SZ);
  unsigned char* o_fg  = (unsigned char*)(base + 28 * SZ);

  taa_init_ws<<<(B_ * N_ + 255) / 256, 256, 0, stream>>>(keys);

  taa_topk_scatter<<<(B_ * M_) / COLS_PER_BLOCK_, 256, 0, stream>>>(
      pred_scores, (const float4*)pred_boxes, (const float4*)gt_boxes,
      gt_mask, keys);

  taa_resolve<<<dim3((N_ + 255) / 256, B_), 256, 0, stream>>>(
      (const float4*)pred_boxes, (const float4*)gt_boxes, gt_labels, keys,
      o_idx, o_lab, o_box, o_sco, o_fg);
}